// WaveletConv2D_4973572129291
// MI455X (gfx1250) — compile-verified
//
#include <hip/hip_runtime.h>

typedef float v2f __attribute__((ext_vector_type(2)));
typedef float v8f __attribute__((ext_vector_type(8)));

// Analysis filters
__constant__ float d_H0[8] = { 0.0322231f, -0.01260397f, -0.09921954f,  0.2978578f,
                               0.80373875f, 0.49761867f, -0.02963553f, -0.07576571f };
__constant__ float d_H1[8] = { 0.07576571f, -0.02963553f, -0.49761867f, 0.80373875f,
                              -0.2978578f, -0.09921954f,  0.01260397f,  0.0322231f };
// Synthesis filters: g0 = reverse(h0), g1 = reverse(h1)
__constant__ float d_G0[8] = {-0.07576571f, -0.02963553f, 0.49761867f, 0.80373875f,
                               0.2978578f, -0.09921954f, -0.01260397f, 0.0322231f };
__constant__ float d_G1[8] = { 0.0322231f,  0.01260397f, -0.09921954f, -0.2978578f,
                               0.80373875f, -0.49761867f, -0.02963553f, 0.07576571f };

// ---------------------------------------------------------------------------
// Horizontal (W) stride-2 8-tap conv, SAME (pad_lo = 3).  (B,H,W,C)->(B,H,W/2,C)
// ---------------------------------------------------------------------------
template<int USE_H1>
__global__ void k_row_dwt(const float* __restrict__ in, float* __restrict__ out,
                          int B, int H, int W, int C) {
  const int Wo = W >> 1;
  const long long total = (long long)B * H * Wo * C;
  long long idx = (long long)blockIdx.x * blockDim.x + threadIdx.x;
  if (idx >= total) return;
  int c  = (int)(idx % C);
  long long t = idx / C;
  int wo = (int)(t % Wo); t /= Wo;
  int h  = (int)(t % H);
  int b  = (int)(t / H);
  const float* F = USE_H1 ? d_H1 : d_H0;
  const float* rowp = in + (((long long)b * H + h) * W) * C + c;
  float acc = 0.f;
  const int wbase = 2 * wo - 3;
#pragma unroll
  for (int k = 0; k < 8; ++k) {
    int w = wbase + k;
    float v = (w >= 0 && w < W) ? rowp[(long long)w * C] : 0.f;
    acc = fmaf(v, F[k], acc);
  }
  out[idx] = acc;
}

// ---------------------------------------------------------------------------
// Vertical (H) stride-2 8-tap conv, SAME (pad_lo = 3).  (B,H,W,C)->(B,H/2,W,C)
// ---------------------------------------------------------------------------
template<int USE_H1>
__global__ void k_col_dwt(const float* __restrict__ in, float* __restrict__ out,
                          int B, int H, int W, int C) {
  const int Ho = H >> 1;
  const long long total = (long long)B * Ho * W * C;
  long long idx = (long long)blockIdx.x * blockDim.x + threadIdx.x;
  if (idx >= total) return;
  int c  = (int)(idx % C);
  long long t = idx / C;
  int w  = (int)(t % W); t /= W;
  int ho = (int)(t % Ho);
  int b  = (int)(t / Ho);
  const float* F = USE_H1 ? d_H1 : d_H0;
  const float* colp = in + ((long long)b * H * W + w) * C + c;
  const long long rowStride = (long long)W * C;
  float acc = 0.f;
  const int hbase = 2 * ho - 3;
#pragma unroll
  for (int k = 0; k < 8; ++k) {
    int h = hbase + k;
    float v = (h >= 0 && h < H) ? colp[(long long)h * rowStride] : 0.f;
    acc = fmaf(v, F[k], acc);
  }
  out[idx] = acc;
}

// ---------------------------------------------------------------------------
// Weight re-layout with K-pair interleave so a WMMA B-fragment is ONE aligned
// 8-byte load per lane:
//   W[i,o,h,w] @ (i*64+o)*1024 + hw   ->   WT[hw][(i>>1)*128 + o*2 + (i&1)]
// grid.x = hw (1024), block = 256, loop over io
// ---------------------------------------------------------------------------
__global__ void k_wt(const float* __restrict__ src, float* __restrict__ dst) {
  const int hw = blockIdx.x;
  for (int io = threadIdx.x; io < 4096; io += blockDim.x) {
    const int i = io >> 6;
    const int o = io & 63;
    dst[(long long)hw * 4096 + (i >> 1) * 128 + o * 2 + (i & 1)] =
        src[(long long)io * 1024 + hw];
  }
}

// ---------------------------------------------------------------------------
// Channel mixing einsum: out[b,hw,o] = sum_i X[b,hw,i] * W_hw[i,o]
// Block = 128 threads = 4 waves; wave w handles o-tile w for one (hw, subband),
// so all waves share X rows and neighboring weight cachelines.
// Each wave: M=32 batch in 2 WMMA tiles, B-fragment shared between them.
// V_WMMA_F32_16X16X4_F32, K = 64 in 16 steps.
// ---------------------------------------------------------------------------
__global__ void __launch_bounds__(128)
k_mix_wmma(const float* __restrict__ sb,   // 4 subbands, each 2097152 f32
           const float* __restrict__ wt,   // 4 relaid weights, each 4194304 f32
           float* __restrict__ outp) {     // 4 outputs, each 2097152 f32
  const int hw   = blockIdx.x;             // 0..1023
  const int s    = blockIdx.y;             // subband 0..3
  const int nt   = threadIdx.x >> 5;       // o-tile 0..3 (one wave each)
  const int lane = threadIdx.x & 31;
  const int half = lane >> 4;              // 0 -> K{0,1}, 1 -> K{2,3}
  const int l15  = lane & 15;

  const float* X  = sb   + (long long)s * 2097152 + (long long)hw * 64;   // [b*65536 + i]
  const float* Wp = wt   + (long long)s * 4194304 + (long long)hw * 4096; // interleaved
  float*       O  = outp + (long long)s * 2097152 + (long long)hw * 64;   // [b*65536 + o]

  v8f acc0 = {};
  v8f acc1 = {};
#pragma unroll
  for (int kc = 0; kc < 16; ++kc) {
    const int i0 = kc * 4 + half * 2;      // even
    // B: 4x16, lanes 0-15 carry K{0,1}, lanes 16-31 carry K{2,3}
    const v2f bf = *(const v2f*)(Wp + (i0 >> 1) * 128 + (nt * 16 + l15) * 2);
    // A: 16x4 per ISA layout (lanes 0-15 rows, VGPR0/1 = K0/K1; 16-31 = K2/K3)
    const v2f a0 = *(const v2f*)(X + (long long)l15 * 65536 + i0);
    const v2f a1 = *(const v2f*)(X + (long long)(l15 + 16) * 65536 + i0);
    acc0 = __builtin_amdgcn_wmma_f32_16x16x4_f32(false, a0, false, bf, (short)0, acc0, false, false);
    acc1 = __builtin_amdgcn_wmma_f32_16x16x4_f32(false, a1, false, bf, (short)0, acc1, false, false);
  }

  // C/D layout: VGPR r -> lanes 0-15: M=r ; lanes 16-31: M=r+8
  const int mofs = (lane < 16) ? 0 : 8;
  const int col  = nt * 16 + l15;
#pragma unroll
  for (int r = 0; r < 8; ++r) {
    O[(long long)(mofs + r) * 65536 + col]      = acc0[r];
    O[(long long)(16 + mofs + r) * 65536 + col] = acc1[r];
  }
}

// ---------------------------------------------------------------------------
// Level-2 synthesis: sum of 4 transposed convs (zero-upsample + 8-tap SAME,
// pad_lo = 3 -> 4x4 effective taps per parity).  (4x(32,32,32,64)) -> (32,64,64,64)
// subband s: vertical filter s<2 ? g0 : g1 ; horizontal filter s&1 ? g1 : g0
// ---------------------------------------------------------------------------
__global__ void k_idwt4(const float* __restrict__ t, float* __restrict__ out) {
  unsigned idx = blockIdx.x * blockDim.x + threadIdx.x;
  if (idx >= 32u * 64u * 64u * 64u) return;
  const int c = idx & 63;
  const int x = (idx >> 6) & 63;
  const int y = (idx >> 12) & 63;
  const int b = idx >> 18;
  const int p = (y + 1) & 1;
  const int q = (x + 1) & 1;
  float acc = 0.f;
#pragma unroll
  for (int a = 0; a < 4; ++a) {
    const int kh = p + 2 * a;
    const int ih = (y - 3 + kh) >> 1;
    if (ih < 0 || ih >= 32) continue;
    const float gv0 = d_G0[kh], gv1 = d_G1[kh];
#pragma unroll
    for (int e = 0; e < 4; ++e) {
      const int kw = q + 2 * e;
      const int iw = (x - 3 + kw) >> 1;
      if (iw < 0 || iw >= 32) continue;
      const float gh0 = d_G0[kw], gh1 = d_G1[kw];
      const long long off = (((long long)b * 32 + ih) * 32 + iw) * 64 + c;
      const float vft = t[off];
      const float vlh = t[2097152 + off];
      const float vhl = t[2 * 2097152 + off];
      const float vhh = t[3 * 2097152 + off];
      acc += gv0 * (gh0 * vft + gh1 * vlh) + gv1 * (gh0 * vhl + gh1 * vhh);
    }
  }
  out[idx] = acc;
}

// ---------------------------------------------------------------------------
// Level-1 synthesis (g0 x g0 only): (32,64,64,64) -> (32,128,128,64)
// ---------------------------------------------------------------------------
__global__ void k_idwt1(const float* __restrict__ res, float* __restrict__ out) {
  unsigned idx = blockIdx.x * blockDim.x + threadIdx.x;
  if (idx >= 32u * 128u * 128u * 64u) return;
  const int c = idx & 63;
  const int x = (idx >> 6) & 127;
  const int y = (idx >> 13) & 127;
  const int b = idx >> 20;
  const int p = (y + 1) & 1;
  const int q = (x + 1) & 1;
  float acc = 0.f;
#pragma unroll
  for (int a = 0; a < 4; ++a) {
    const int kh = p + 2 * a;
    const int ih = (y - 3 + kh) >> 1;
    if (ih < 0 || ih >= 64) continue;
    const float gv = d_G0[kh];
#pragma unroll
    for (int e = 0; e < 4; ++e) {
      const int kw = q + 2 * e;
      const int iw = (x - 3 + kw) >> 1;
      if (iw < 0 || iw >= 64) continue;
      acc = fmaf(gv * d_G0[kw],
                 res[(((long long)b * 64 + ih) * 64 + iw) * 64 + c], acc);
    }
  }
  out[idx] = acc;
}

// ---------------------------------------------------------------------------
extern "C" void kernel_launch(void* const* d_in, const int* in_sizes, int n_in,
                              void* d_out, int out_size, void* d_ws, size_t ws_size,
                              hipStream_t stream) {
  const float* x  = (const float*)d_in[0];
  const float* w1 = (const float*)d_in[1];
  const float* w2 = (const float*)d_in[2];
  const float* w3 = (const float*)d_in[3];
  const float* w4 = (const float*)d_in[4];
  float* out = (float*)d_out;
  float* ws  = (float*)d_ws;

  // Workspace arena (floats). Total 58,720,256 f32 = 224 MiB.
  float* TMP1 = ws;                   // (32,128,64,64)  16,777,216 — dead after APP1
  float* APP1 = ws + 16777216;        // (32,64,64,64)    8,388,608
  float* T2L  = ws + 25165824;        // (32,64,32,64)    4,194,304
  float* T2H  = ws + 29360128;        // (32,64,32,64)    4,194,304
  float* SB   = ws + 33554432;        // 4 x (32,32,32,64) 8,388,608  [ft,lh,hl,hh]
  float* OUT4 = ws + 41943040;        // 4 x (32,32,32,64) 8,388,608
  float* RES  = ws + 50331648;        // (32,64,64,64)    8,388,608
  float* WT   = TMP1;                 // 4 x 4,194,304 — reuses TMP1 (dead by then)

  const int TB = 256;

  // Level 1 (only the approx band is ever used downstream)
  k_row_dwt<0><<<dim3(16777216 / TB), TB, 0, stream>>>(x, TMP1, 32, 128, 128, 64);
  k_col_dwt<0><<<dim3(8388608 / TB), TB, 0, stream>>>(TMP1, APP1, 32, 128, 64, 64);

  // Level 2: shared row passes, then 4 column passes
  k_row_dwt<0><<<dim3(4194304 / TB), TB, 0, stream>>>(APP1, T2L, 32, 64, 64, 64);
  k_row_dwt<1><<<dim3(4194304 / TB), TB, 0, stream>>>(APP1, T2H, 32, 64, 64, 64);
  k_col_dwt<0><<<dim3(2097152 / TB), TB, 0, stream>>>(T2L, SB,               32, 64, 32, 64); // ft  (h0,h0)
  k_col_dwt<0><<<dim3(2097152 / TB), TB, 0, stream>>>(T2H, SB + 2097152,     32, 64, 32, 64); // lh  (h0 v, h1 h)
  k_col_dwt<1><<<dim3(2097152 / TB), TB, 0, stream>>>(T2L, SB + 2 * 2097152, 32, 64, 32, 64); // hl  (h1 v, h0 h)
  k_col_dwt<1><<<dim3(2097152 / TB), TB, 0, stream>>>(T2H, SB + 3 * 2097152, 32, 64, 32, 64); // hh  (h1,h1)

  // Weight re-layout for single-b64 WMMA B-fragments (after TMP1 is dead)
  k_wt<<<dim3(1024), TB, 0, stream>>>(w1, WT);
  k_wt<<<dim3(1024), TB, 0, stream>>>(w2, WT + 4194304);
  k_wt<<<dim3(1024), TB, 0, stream>>>(w3, WT + 2 * 4194304);
  k_wt<<<dim3(1024), TB, 0, stream>>>(w4, WT + 3 * 4194304);

  // Per-position 64x64 channel mixing on the matrix pipe (fp32 WMMA)
  k_mix_wmma<<<dim3(1024, 4), 128, 0, stream>>>(SB, WT, OUT4);

  // Inverse transform
  k_idwt4<<<dim3(8388608 / TB), TB, 0, stream>>>(OUT4, RES);
  k_idwt1<<<dim3(33554432 / TB), TB, 0, stream>>>(RES, out);
}